// RankRTModelA_39273180954762
// MI455X (gfx1250) — compile-verified
//
#include <hip/hip_runtime.h>
#include <math.h>

// Problem constants from the reference
#define N_STIM1    21   // N_STIM + 1 table rows
#define N_DIM      3
#define TAB_STRIDE 4    // pad table rows to 16B: gathers become ds_load_b96, bank-friendly
#define BLOCK      256  // 8 wave32s

// ---- fast trans helpers: single V_LOG_F32 / V_EXP_F32 / V_RCP_F32 ops ----
#define LOG2E 1.4426950408889634f
#define LN2   0.6931471805599453f

__device__ __forceinline__ float fast_log2(float x) { return __builtin_amdgcn_logf(x); }
__device__ __forceinline__ float fast_exp2(float x) { return __builtin_amdgcn_exp2f(x); }
__device__ __forceinline__ float fast_rcp (float x) { return __builtin_amdgcn_rcpf(x); }
// x >= 0, y > 0:  x^y = exp2(y*log2 x); log2(0)=-inf -> exp2(-inf)=0 (matches pow)
__device__ __forceinline__ float fast_pow (float x, float y) { return fast_exp2(y * fast_log2(x)); }
__device__ __forceinline__ float fast_exp (float x) { return fast_exp2(x * LOG2E); }
__device__ __forceinline__ float fast_ln  (float x) { return fast_log2(x) * LN2; }

__global__ __launch_bounds__(BLOCK) void rankrt_kernel(
    const int*   __restrict__ stim,     // (B,5) int32, values 1..20
    const float* __restrict__ table,    // (21,3) f32
    const float* __restrict__ w,        // (3,)
    const float* __restrict__ rho_p,
    const float* __restrict__ beta_p,
    const float* __restrict__ tau_p,
    const float* __restrict__ gamma_p,
    const float* __restrict__ upper_p,
    const float* __restrict__ mid_p,
    const float* __restrict__ rate_p,
    float* __restrict__ out_rank,       // (B,4)
    float* __restrict__ out_rt,         // (B,)
    int B)
{
    __shared__ int   sh_stim[BLOCK * 5];            // 5120 B: block's stimulus tile
    __shared__ float tab[N_STIM1 * TAB_STRIDE];     // 336 B: padded lookup table

    const int tid  = threadIdx.x;
    const int row0 = blockIdx.x * BLOCK;

    // ----------------------------------------------------------------------
    // Stage ALL inputs through the CDNA5 async global->LDS engine (ASYNCcnt).
    // ----------------------------------------------------------------------
    // (a) Stimulus tile: nrows*5 dwords, bulk as coalesced B128 (16B/lane),
    //     leftover dwords as B32.
    {
        int nrows = B - row0; if (nrows > BLOCK) nrows = BLOCK;
        const int ndw   = nrows * 5;                 // dwords in this block's tile
        int nb128 = ndw >> 2; if (nb128 > BLOCK) nb128 = BLOCK;
        const int start = nb128 << 2;                // dwords covered by B128 part
        const int cnt   = ndw - start;               // tail dwords (<= BLOCK)

        const int* gbase = stim + (size_t)row0 * 5;  // 16B-aligned (5120B per block)

        if (tid < nb128) {
            unsigned lds_dst = (unsigned)(size_t)(const void*)&sh_stim[tid * 4];
            unsigned goff    = (unsigned)(tid * 16);
            asm volatile("global_load_async_to_lds_b128 %0, %1, %2"
                         :: "v"(lds_dst), "v"(goff), "s"(gbase) : "memory");
        }
        if (tid < cnt) {
            unsigned lds_dst = (unsigned)(size_t)(const void*)&sh_stim[start + tid];
            unsigned goff    = (unsigned)((start + tid) * 4);
            asm volatile("global_load_async_to_lds_b32 %0, %1, %2"
                         :: "v"(lds_dst), "v"(goff), "s"(gbase) : "memory");
        }
    }
    // (b) 21x3 table, scattered into the padded (stride-4) LDS layout.
    if (tid < N_STIM1 * N_DIM) {
        const int row = tid / N_DIM;
        const int col = tid - row * N_DIM;
        unsigned lds_dst = (unsigned)(size_t)(const void*)&tab[row * TAB_STRIDE + col];
        unsigned goff    = (unsigned)(tid * 4);
        asm volatile("global_load_async_to_lds_b32 %0, %1, %2"
                     :: "v"(lds_dst), "v"(goff), "s"(table) : "memory");
    }
    // Issuing waves drain ASYNCcnt before the barrier; no-op for the rest.
    asm volatile("s_wait_asynccnt 0x0" ::: "memory");
    __syncthreads();

    const int i = row0 + tid;
    if (i >= B) return;

    // Uniform parameters (scalar loads)
    const float w0 = w[0], w1 = w[1], w2 = w[2];
    const float rho      = rho_p[0];
    const float beta     = beta_p[0];
    const float tau      = tau_p[0];
    const float gamma    = gamma_p[0];
    const float upper    = upper_p[0];
    const float midpoint = mid_p[0];
    const float rate     = rate_p[0];
    const float tau_over_rho = tau * fast_rcp(rho);  // fold d^tau = S^(tau/rho)

    // Per-row indices from LDS (lane t -> bank 5t mod 64; gcd(5,64)=1 => no conflicts)
    const int* my = &sh_stim[tid * 5];
    const int q  = my[0];
    int ridx[4];
    ridx[0] = my[1]; ridx[1] = my[2]; ridx[2] = my[3]; ridx[3] = my[4];

    // Query embedding (ds_load_b96)
    const float* zq = &tab[q * TAB_STRIDE];
    const float zq0 = zq[0], zq1 = zq[1], zq2 = zq[2];

    float s[4];
    float ssum = 0.0f;
#pragma unroll
    for (int j = 0; j < 4; ++j) {
        const float* zr = &tab[ridx[j] * TAB_STRIDE];
        const float dx = fabsf(zq0 - zr[0]);
        const float dy = fabsf(zq1 - zr[1]);
        const float dz = fabsf(zq2 - zr[2]);
        // S = sum_k w_k |dz_k|^rho ; s = exp(-beta * S^(tau/rho)) + gamma
        const float S  = w0 * fast_pow(dx, rho)
                       + w1 * fast_pow(dy, rho)
                       + w2 * fast_pow(dz, rho);
        const float sj = fast_exp(-beta * fast_pow(S, tau_over_rho)) + gamma;
        s[j] = sj;
        ssum += sj;
    }

    const float inv = fast_rcp(ssum);
    float entropy = 0.0f;
    float4 rank;
    {
        const float r0 = s[0] * inv, r1 = s[1] * inv,
                    r2 = s[2] * inv, r3 = s[3] * inv;
        entropy -= r0 * fast_ln(r0);
        entropy -= r1 * fast_ln(r1);
        entropy -= r2 * fast_ln(r2);
        entropy -= r3 * fast_ln(r3);
        rank.x = r0; rank.y = r1; rank.z = r2; rank.w = r3;
    }

    // Aligned 16B vector store for the rank row (global_store_b128)
    reinterpret_cast<float4*>(out_rank)[i] = rank;
    out_rt[i] = upper * fast_rcp(1.0f + fast_exp(-rate * (entropy - midpoint)));
}

extern "C" void kernel_launch(void* const* d_in, const int* in_sizes, int n_in,
                              void* d_out, int out_size, void* d_ws, size_t ws_size,
                              hipStream_t stream) {
    (void)n_in; (void)out_size; (void)d_ws; (void)ws_size;

    const int*   stim   = (const int*)  d_in[0];
    const float* table  = (const float*)d_in[1];
    const float* w      = (const float*)d_in[2];
    const float* rho    = (const float*)d_in[3];
    const float* beta   = (const float*)d_in[4];
    const float* tau    = (const float*)d_in[5];
    const float* gamma  = (const float*)d_in[6];
    const float* upper  = (const float*)d_in[7];
    const float* mid    = (const float*)d_in[8];
    const float* rate   = (const float*)d_in[9];

    const int B = in_sizes[0] / 5;

    float* out_rank = (float*)d_out;                 // (B,4) flat first
    float* out_rt   = out_rank + (size_t)B * 4;      // then (B,1)

    const int grid = (B + BLOCK - 1) / BLOCK;
    rankrt_kernel<<<grid, BLOCK, 0, stream>>>(
        stim, table, w, rho, beta, tau, gamma, upper, mid, rate,
        out_rank, out_rt, B);
}